// GCNConv_dgl_attn_31078383353909
// MI455X (gfx1250) — compile-verified
//
#include <hip/hip_runtime.h>

typedef float v2f __attribute__((ext_vector_type(2)));
typedef float v8f __attribute__((ext_vector_type(8)));

#define D_IN  256
#define D_OUT 128

// ---------------------------------------------------------------------------
// Kernel 1: h[n][o] = sum_i x[n][i] * W[o][i] + b[o]   (fp32 WMMA 16x16x4)
// Block = 256 threads = 8 waves. Block handles one 16-row tile of x/h.
// Wave w computes the 16x16 output tile at columns [16w, 16w+16).
// ---------------------------------------------------------------------------
__global__ __launch_bounds__(256) void gcn_gemm_wmma(
    const float* __restrict__ x,   // [N, 256]
    const float* __restrict__ W,   // [128, 256]
    const float* __restrict__ b,   // [128]
    float* __restrict__ h,         // [N, 128]
    int nNodes) {
  const int lane    = threadIdx.x & 31;
  const int wave    = threadIdx.x >> 5;          // N-tile index 0..7
  const int rowTile = blockIdx.x;                // 16-row tile of nodes

  const int m16   = lane & 15;                   // row within tile (A) / col (B)
  const int khalf = (lane >> 4) << 1;            // 0 for lanes 0-15, 2 for 16-31

  int row = rowTile * 16 + m16;
  if (row >= nNodes) row = nNodes - 1;           // clamp (exact division anyway)
  const int col = wave * 16 + m16;               // output column 0..127

  const float* __restrict__ xrow = x + (size_t)row * D_IN + khalf;
  const float* __restrict__ wrow = W + (size_t)col * D_IN + khalf;

  v8f acc = {};
#pragma unroll
  for (int k = 0; k < D_IN; k += 4) {
    // A 16x4 fp32: lane m holds K=khalf+0 (v0), K=khalf+1 (v1)
    v2f a = *(const v2f*)(xrow + k);
    // B 4x16 fp32: VGPR0 = row K=khalf, VGPR1 = row K=khalf+1; B[k][n] = W[n][k]
    v2f bm = *(const v2f*)(wrow + k);
    acc = __builtin_amdgcn_wmma_f32_16x16x4_f32(
        /*neg_a=*/false, a, /*neg_b=*/false, bm,
        /*c_mod=*/(short)0, acc, /*reuse_a=*/false, /*reuse_b=*/false);
  }

  // C/D layout: VGPR r, lanes 0-15 -> M=r, N=lane; lanes 16-31 -> M=r+8, N=lane-16
  const float bias  = b[col];
  const int   mBase = rowTile * 16 + ((lane >> 4) << 3);   // +0 or +8
#pragma unroll
  for (int r = 0; r < 8; ++r) {
    const int m = mBase + r;
    if (m < nNodes) h[(size_t)m * D_OUT + col] = acc[r] + bias;
  }
}

// ---------------------------------------------------------------------------
// Kernel 2: out[v] = sum_{e: dst[e]==v} h[src[e]] * w[e]
// dst is sorted -> one wave per node, edge range via binary search, no atomics.
// Lane l accumulates a float4 for columns 4l..4l+3 (32*4 = 128 = D_OUT).
// h (51 MB) stays L2-resident on MI455X (192 MB L2), so the 819 MB of edge
// gather traffic is served from L2, not HBM.
// ---------------------------------------------------------------------------
__device__ __forceinline__ int lower_bound_i32(const int* __restrict__ a,
                                               int n, int key) {
  int lo = 0, hi = n;
  while (lo < hi) {
    int mid = (lo + hi) >> 1;
    if (a[mid] < key) lo = mid + 1; else hi = mid;
  }
  return lo;
}

__global__ __launch_bounds__(256) void gcn_scatter_sum(
    const float* __restrict__ h,    // [N, 128]
    const int* __restrict__ src,    // [E]
    const int* __restrict__ dst,    // [E] sorted ascending
    const float* __restrict__ w,    // [E]
    float* __restrict__ out,        // [N, 128]
    int nNodes, int nEdges) {
  const int lane = threadIdx.x & 31;
  const int node = blockIdx.x * (blockDim.x >> 5) + (threadIdx.x >> 5);
  if (node >= nNodes) return;

  // Edge range [lo, hi) with dst == node (uniform across the wave).
  const int lo = lower_bound_i32(dst, nEdges, node);
  const int hi = lower_bound_i32(dst, nEdges, node + 1);

  const int c = lane << 2;  // column base for this lane
  float4 acc = make_float4(0.f, 0.f, 0.f, 0.f);

  for (int e = lo; e < hi; ++e) {
    const int   s  = src[e];   // uniform per wave
    const float we = w[e];     // uniform per wave
    const float4 hv = *(const float4*)(h + (size_t)s * D_OUT + c);
    acc.x += hv.x * we;
    acc.y += hv.y * we;
    acc.z += hv.z * we;
    acc.w += hv.w * we;
  }

  // Each node written exactly once -> plain coalesced store, zero-fills
  // isolated nodes (output buffer is poisoned by the harness).
  *(float4*)(out + (size_t)node * D_OUT + c) = acc;
}

// ---------------------------------------------------------------------------
extern "C" void kernel_launch(void* const* d_in, const int* in_sizes, int n_in,
                              void* d_out, int out_size, void* d_ws, size_t ws_size,
                              hipStream_t stream) {
  const float* x   = (const float*)d_in[0];  // [N, 256] fp32
  const int*   src = (const int*)  d_in[1];  // [E] int32
  const int*   dst = (const int*)  d_in[2];  // [E] int32, sorted
  const float* w   = (const float*)d_in[3];  // [E] fp32
  const float* W   = (const float*)d_in[4];  // [128, 256] fp32
  const float* b   = (const float*)d_in[5];  // [128] fp32

  float* out = (float*)d_out;
  float* h   = (float*)d_ws;                 // [N, 128] fp32 scratch (51.2 MB)

  const int nNodes = in_sizes[0] / D_IN;
  const int nEdges = in_sizes[1];

  // Phase 1: dense GEMM via fp32 WMMA.
  const int rowTiles = (nNodes + 15) / 16;
  gcn_gemm_wmma<<<rowTiles, 256, 0, stream>>>(x, W, b, h, nNodes);

  // Phase 2: atomic-free segmented reduction, one wave per destination node.
  const int wavesPerBlock = 256 / 32;
  const int blocks = (nNodes + wavesPerBlock - 1) / wavesPerBlock;
  gcn_scatter_sum<<<blocks, 256, 0, stream>>>(h, src, dst, w, out, nNodes, nEdges);
}